// DetectionOutput_64407329571002
// MI455X (gfx1250) — compile-verified
//
#include <hip/hip_runtime.h>
#include <math.h>

// ---------------- problem constants (match reference) ----------------
#define B_       32
#define P_       250000
#define TOPK_    200
#define NCHUNK_  16
#define CHUNK_   (P_ / NCHUNK_)          // 15625 (exact)
#define NIT_     ((CHUNK_ + 255) / 256)  // 62 tiles of 256 priors
#define KLOC_    256                     // per-chunk kept candidates (>= TOPK)
#define NCAND_   (NCHUNK_ * KLOC_)       // 4096 merge candidates per batch
#define CONF_TH  0.01f
#define NMS_TH   0.45f

typedef unsigned long long u64;

// ---------------- Tensor Data Mover (CDNA5 TDM) ----------------
typedef unsigned int tdm_u32x4 __attribute__((ext_vector_type(4)));
typedef int          tdm_i32x8 __attribute__((ext_vector_type(8)));
typedef int          tdm_i32x4 __attribute__((ext_vector_type(4)));

__device__ __forceinline__ unsigned uni32(unsigned x) {
  return (unsigned)__builtin_amdgcn_readfirstlane((int)x);
}

// 1-D tile DMA: `elems` elements of (1<<dsz_code) bytes, global -> LDS.
// D# group0: count=1 | lds_addr | global_addr[56:0] | type=2
// D# group1: data_size | tensor_dim0=elems | tensor_dim1=1 | tile_dim0=elems |
//            tile_dim1=1 | dim0_stride=elems
__device__ __forceinline__ void tdm_load_1d(const void* gaddr, unsigned lds_off,
                                            unsigned elems, unsigned dsz_code) {
  unsigned long long ga = (unsigned long long)(uintptr_t)gaddr;
  tdm_u32x4 g0;
  g0[0] = uni32(1u);                                   // count=1 (valid), user mode
  g0[1] = uni32(lds_off);                              // LDS byte address
  g0[2] = uni32((unsigned)(ga & 0xFFFFFFFFull));       // global addr lo
  g0[3] = uni32(((unsigned)((ga >> 32) & 0x1FFFFFFull)) | 0x80000000u); // hi | type=2
  tdm_i32x8 g1;
  g1[0] = (int)uni32(dsz_code << 16);                  // data_size code
  g1[1] = (int)uni32((elems & 0xFFFFu) << 16);         // tensor_dim0[15:0]
  g1[2] = (int)uni32((elems >> 16) | (1u << 16));      // tensor_dim0[31:16] | tensor_dim1=1
  g1[3] = (int)uni32((elems & 0xFFFFu) << 16);         // tile_dim0 (0 => NOP)
  g1[4] = (int)uni32(1u);                              // tile_dim1=1, tile_dim2=0
  g1[5] = (int)uni32(elems);                           // tensor_dim0_stride lo32
  g1[6] = 0;
  g1[7] = 0;
  tdm_i32x4 gz4; gz4[0] = 0; gz4[1] = 0; gz4[2] = 0; gz4[3] = 0;
  tdm_i32x8 gz8;
  gz8[0] = 0; gz8[1] = 0; gz8[2] = 0; gz8[3] = 0;
  gz8[4] = 0; gz8[5] = 0; gz8[6] = 0; gz8[7] = 0;
  // clang-23 / therock-10.0 form: 6 args (g0, g1, g2, g3, extra, cpol)
  __builtin_amdgcn_tensor_load_to_lds(g0, g1, gz4, gz4, gz8, 0);
}

// =====================================================================
// Stage 1: per (batch, chunk) block of 256 threads keeps running top-256
// of 15625 priors by 64-bit key = (ordered(conf1-conf0) << 32) | ~prior.
// conf is streamed through LDS with per-wave double-buffered TDM DMAs.
// =====================================================================
__global__ __launch_bounds__(256)
void detout_stage1_topk(const float* __restrict__ conf, u64* __restrict__ cand) {
  __shared__ u64   best[KLOC_];
  __shared__ u64   tile[KLOC_];
  __shared__ float confbuf[2][512];   // 256 priors x 2 classes, double buffered
  __shared__ int   sflag;

  const int tid  = threadIdx.x;
  const int b    = blockIdx.y;
  const int ck   = blockIdx.x;
  const int w    = tid >> 5;          // wave id (8 waves, wave32)
  const int lane = tid & 31;
  const long chunkStart = (long)ck * CHUNK_;
  const float* confB = conf + (long long)b * P_ * 2;

  best[tid] = 0ull;
  __syncthreads();

  // prologue: each wave DMAs its 32-prior (64-float) slice of tile 0
  {
    long p0  = chunkStart + (long)w * 32;
    long rem = ((long)P_ - p0) * 2;
    unsigned e = (unsigned)(rem < 0 ? 0 : (rem > 64 ? 64 : rem));
    long pc  = (p0 < (long)P_) ? p0 : 0;
    tdm_load_1d(confB + pc * 2, (unsigned)(uintptr_t)&confbuf[0][w * 64], e, 2u);
  }

  for (int i = 0; i < NIT_; ++i) {
    if (i + 1 < NIT_) {   // issue next tile, then wait for current (in-order TDM)
      long p0  = chunkStart + (long)(i + 1) * 256 + (long)w * 32;
      long rem = ((long)P_ - p0) * 2;
      unsigned e = (unsigned)(rem < 0 ? 0 : (rem > 64 ? 64 : rem));
      long pc  = (p0 < (long)P_) ? p0 : 0;
      tdm_load_1d(confB + pc * 2, (unsigned)(uintptr_t)&confbuf[(i + 1) & 1][w * 64], e, 2u);
      __builtin_amdgcn_s_wait_tensorcnt(1);
    } else {
      __builtin_amdgcn_s_wait_tensorcnt(0);
    }
    __asm__ __volatile__("" ::: "memory");

    // key for this lane's prior (each wave reads only LDS its own TDM wrote)
    const int pl = i * 256 + w * 32 + lane;
    volatile const float* cb = &confbuf[i & 1][0];
    float c0 = cb[w * 64 + lane * 2 + 0];
    float c1 = cb[w * 64 + lane * 2 + 1];
    float d  = c1 - c0;                               // monotone in softmax score
    unsigned fb  = __float_as_uint(d);
    unsigned ord = (fb & 0x80000000u) ? ~fb : (fb | 0x80000000u);
    unsigned p   = (unsigned)(chunkStart + pl);
    u64 key = ((u64)ord << 32) | (u64)(~p);           // lower index wins ties
    if (pl >= CHUNK_) key = 0ull;

    tile[tid] = key;
    if (tid == 0) sflag = 0;
    __syncthreads();
    if (key > best[KLOC_ - 1]) sflag = 1;             // beats current min?
    __syncthreads();

    if (sflag) {                                      // block-uniform branch
      // bitonic sort tile[] descending (256 elements, 256 threads)
      for (int k = 2; k <= KLOC_; k <<= 1) {
        for (int j = k >> 1; j > 0; j >>= 1) {
          int pr = tid ^ j;
          if (pr > tid) {
            u64 x = tile[tid], y = tile[pr];
            bool desc = ((tid & k) == 0);
            if (desc ? (x < y) : (x > y)) { tile[tid] = y; tile[pr] = x; }
          }
          __syncthreads();
        }
      }
      // top-K merge: max(best[i], tile[K-1-i]) is bitonic and contains top-K
      u64 a = best[tid];
      u64 c = tile[KLOC_ - 1 - tid];
      u64 m = a > c ? a : c;
      __syncthreads();
      best[tid] = m;
      __syncthreads();
      for (int j = KLOC_ >> 1; j > 0; j >>= 1) {      // bitonic half-cleaners
        int pr = tid ^ j;
        if (pr > tid) {
          u64 x = best[tid], y = best[pr];
          if (x < y) { best[tid] = y; best[pr] = x; }
        }
        __syncthreads();
      }
    }
  }
  cand[((long long)b * NCHUNK_ + ck) * KLOC_ + tid] = best[tid];
}

// =====================================================================
// Stage 2: per batch, sort 4096 candidates (TDM-loaded into LDS), take
// top-200, decode boxes, greedy NMS, write [2,200,5] output plane.
// =====================================================================
__global__ __launch_bounds__(1024)
void detout_stage2_nms(const float* __restrict__ loc, const float* __restrict__ priors,
                       const u64* __restrict__ cand, float* __restrict__ out) {
  __shared__ u64   arr[NCAND_];       // 32 KB
  __shared__ float sc[TOPK_];
  __shared__ float bx[TOPK_][4];
  __shared__ int   vld[TOPK_];
  __shared__ int   keep[TOPK_];
  __shared__ int   supp;

  const int tid = threadIdx.x;
  const int b   = blockIdx.x;
  const int w   = tid >> 5;           // 32 waves

  // each wave DMAs its 128-u64 (1 KB) slice of the candidate table
  tdm_load_1d(cand + (long long)b * NCAND_ + w * 128,
              (unsigned)(uintptr_t)&arr[w * 128], 128u, 3u);
  __builtin_amdgcn_s_wait_tensorcnt(0);
  __asm__ __volatile__("" ::: "memory");
  __syncthreads();

  // bitonic sort 4096 descending (1024 threads, 4 elems each)
  for (int k = 2; k <= NCAND_; k <<= 1) {
    for (int j = k >> 1; j > 0; j >>= 1) {
      for (int i0 = tid; i0 < NCAND_; i0 += 1024) {
        int pr = i0 ^ j;
        if (pr > i0) {
          u64 x = arr[i0], y = arr[pr];
          bool desc = ((i0 & k) == 0);
          if (desc ? (x < y) : (x > y)) { arr[i0] = y; arr[pr] = x; }
        }
      }
      __syncthreads();
    }
  }

  // decode top-200 (exact score recovered from key bits; gather loc/priors)
  if (tid < TOPK_) {
    u64 key      = arr[tid];
    unsigned ord = (unsigned)(key >> 32);
    unsigned idx = ~((unsigned)key);
    unsigned fb  = (ord & 0x80000000u) ? (ord & 0x7FFFFFFFu) : ~ord;
    float d = __uint_as_float(fb);
    float s = 1.0f / (1.0f + expf(-d));               // softmax[:,1] for 2 classes

    const float* L  = loc + (((long long)b * P_) + idx) * 4;
    const float* Pr = priors + (long long)idx * 4;
    float px0 = Pr[0], py0 = Pr[1], px1 = Pr[2], py1 = Pr[3];
    float pw = px1 - px0, ph = py1 - py0;
    float pcx = (px0 + px1) * 0.5f, pcy = (py0 + py1) * 0.5f;
    float cx = pcx + L[0] * pw;
    float cy = pcy + L[1] * ph;
    float wv = pw * expf(L[2]);
    float hv = ph * expf(L[3]);
    bx[tid][0] = cx - wv * 0.5f;
    bx[tid][1] = cy - hv * 0.5f;
    bx[tid][2] = cx + wv * 0.5f;
    bx[tid][3] = cy + hv * 0.5f;
    sc[tid]  = s;
    vld[tid] = (s > CONF_TH) ? 1 : 0;
    keep[tid] = 0;
  }
  __syncthreads();

  // greedy hard-NMS, sequential in i, parallel suppression test over j<i
  for (int i = 0; i < TOPK_; ++i) {
    if (tid == 0) supp = 0;
    __syncthreads();
    if (tid < i && keep[tid]) {
      float lx = fmaxf(bx[i][0], bx[tid][0]);
      float ly = fmaxf(bx[i][1], bx[tid][1]);
      float rx = fminf(bx[i][2], bx[tid][2]);
      float ry = fminf(bx[i][3], bx[tid][3]);
      float inter = fmaxf(rx - lx, 0.0f) * fmaxf(ry - ly, 0.0f);
      float ai = fmaxf(bx[i][2] - bx[i][0], 0.0f) * fmaxf(bx[i][3] - bx[i][1], 0.0f);
      float aj = fmaxf(bx[tid][2] - bx[tid][0], 0.0f) * fmaxf(bx[tid][3] - bx[tid][1], 0.0f);
      float iou = inter / fmaxf(ai + aj - inter, 1e-9f);
      if (iou > NMS_TH) supp = 1;
    }
    __syncthreads();
    if (tid == 0) keep[i] = vld[i] && !supp;
    __syncthreads();
  }

  // write output [B,2,TOPK,5]: class 0 zeroed, class 1 = (score, box) * keep
  if (tid < TOPK_) {
    float kf = keep[tid] ? 1.0f : 0.0f;
    long long o1 = (((long long)b * 2 + 1) * TOPK_ + tid) * 5;
    out[o1 + 0] = sc[tid] * kf;
    out[o1 + 1] = bx[tid][0] * kf;
    out[o1 + 2] = bx[tid][1] * kf;
    out[o1 + 3] = bx[tid][2] * kf;
    out[o1 + 4] = bx[tid][3] * kf;
    long long o0 = (((long long)b * 2 + 0) * TOPK_ + tid) * 5;
    out[o0 + 0] = 0.0f; out[o0 + 1] = 0.0f; out[o0 + 2] = 0.0f;
    out[o0 + 3] = 0.0f; out[o0 + 4] = 0.0f;
  }
}

// =====================================================================
extern "C" void kernel_launch(void* const* d_in, const int* in_sizes, int n_in,
                              void* d_out, int out_size, void* d_ws, size_t ws_size,
                              hipStream_t stream) {
  (void)in_sizes; (void)n_in; (void)out_size; (void)ws_size;
  const float* loc    = (const float*)d_in[0];   // [B,P,4]
  const float* conf   = (const float*)d_in[1];   // [B,P,2]
  const float* priors = (const float*)d_in[2];   // [P,4]
  float* out = (float*)d_out;                    // [B,2,TOPK,5]
  u64*   cand = (u64*)d_ws;                      // [B, NCHUNK, 256] = 1 MB

  dim3 g1(NCHUNK_, B_);
  detout_stage1_topk<<<g1, 256, 0, stream>>>(conf, cand);
  detout_stage2_nms<<<B_, 1024, 0, stream>>>(loc, priors, cand, out);
}